// PolymorphicSNN_17901423690366
// MI455X (gfx1250) — compile-verified
//
#include <hip/hip_runtime.h>
#include <hip/hip_bf16.h>

// ---------------------------------------------------------------------------
// PolymorphicSNN on MI455X (gfx1250, wave32).
//
// Exact-math reduction of the reference:
//   reg_out = heaviside(x @ W_lin.T + b_lin - 1)  in {0,1}
//   spk_mode = heaviside(reg_out - 1) == 0  ALWAYS  -> mixed == 0 exactly.
// Output [B, 8448] = [reg_out (256 cols) | zeros (8192 cols)].
//
// Kernel 1: f32 WMMA GEMM (V_WMMA_F32_16X16X4_F32) + bias + threshold,
//           writes cols 0..255.
// Kernel 2: float4 zero-fill of cols 256..8447 (dominant cost: ~277 MB
//           of stores ~ 12 us at 23.3 TB/s HBM).
// ---------------------------------------------------------------------------

typedef float v2f __attribute__((ext_vector_type(2)));
typedef float v8f __attribute__((ext_vector_type(8)));

#define N_IN        256
#define P_SEL       32
#define OUT_COLS    (N_IN * (1 + P_SEL))   // 8448
#define LDS_PITCH   260                    // 256 + 4 pad: breaks LDS bank conflicts
#define THRESHOLD_F 1.0f

// GEMM: grid = (2, B/16), block = 256 threads (8 waves).
// Block computes a 16-row x 128-col slab; each wave one 16x16 tile.
__global__ __launch_bounds__(256)
void snn_gemm_heaviside(const float* __restrict__ x,
                        const float* __restrict__ W,   // [256,256] row-major, h=x@W.T
                        const float* __restrict__ bias,
                        float* __restrict__ out)
{
    __shared__ float xs[16 * LDS_PITCH];

    const int tid   = threadIdx.x;
    const int wave  = tid >> 5;          // 0..7 (wave32)
    const int lane  = tid & 31;
    const int row0  = blockIdx.y * 16;   // output row tile
    const int n0    = blockIdx.x * 128 + wave * 16;  // output col tile (per wave)

    // Stage the 16x256 x-tile into LDS (coalesced, 16 iters of 256 threads).
    #pragma unroll
    for (int idx = tid; idx < 16 * N_IN; idx += 256) {
        const int r = idx >> 8;
        const int c = idx & (N_IN - 1);
        xs[r * LDS_PITCH + c] = x[(size_t)(row0 + r) * N_IN + c];
    }
    __syncthreads();

    // WMMA f32 16x16x4 operand addressing (ISA 7.12.2):
    //   A (16x4): lane%16 = M; K = 2*(lane/16) + {vgpr0,vgpr1}
    //   B (4x16): lane%16 = N; K = 2*(lane/16) + {vgpr0,vgpr1}
    //   C/D     : lane%16 = N; M = vgpr + 8*(lane/16)
    const int lane16 = lane & 15;
    const int koff   = (lane >> 4) * 2;

    const float* arow = &xs[lane16 * LDS_PITCH + koff];                 // A: x row
    const float* wrow = W + (size_t)(n0 + lane16) * N_IN + koff;        // B: W_lin row (B[k][n] = W[n][k])

    v8f acc = {0.f, 0.f, 0.f, 0.f, 0.f, 0.f, 0.f, 0.f};

    #pragma unroll 8
    for (int kk = 0; kk < N_IN; kk += 4) {
        v2f a = *(const v2f*)(arow + kk);   // 8B-aligned LDS load
        v2f b = *(const v2f*)(wrow + kk);   // 8B-aligned global load
        // (neg_a, A, neg_b, B, c_mod, C, reuse_a, reuse_b)
        acc = __builtin_amdgcn_wmma_f32_16x16x4_f32(
                  false, a, false, b, (short)0, acc, false, false);
    }

    // Epilogue: bias + heaviside(h - 1), scattered per the C/D layout.
    const int   col   = n0 + lane16;
    const float bcol  = bias[col];
    const int   mbase = (lane >> 4) * 8;
    #pragma unroll
    for (int r = 0; r < 8; ++r) {
        const int   m = mbase + r;
        const float h = acc[r] + bcol;
        out[(size_t)(row0 + m) * OUT_COLS + col] =
            (h - THRESHOLD_F > 0.0f) ? 1.0f : 0.0f;
    }
}

// Zero-fill cols 256..8447 of every row with float4 stores (grid-stride).
__global__ __launch_bounds__(256)
void snn_zero_tail(float4* __restrict__ out4, long long total4)
{
    const float4 z = make_float4(0.f, 0.f, 0.f, 0.f);
    long long i      = (long long)blockIdx.x * blockDim.x + threadIdx.x;
    long long stride = (long long)gridDim.x * blockDim.x;
    // Per row: 8192 zero floats = 2048 float4; row pitch 8448 floats = 2112
    // float4; zeros start 64 float4 into the row.
    for (; i < total4; i += stride) {
        const long long row = i >> 11;          // /2048
        const long long c4  = i & 2047;         // %2048
        out4[row * 2112 + 64 + c4] = z;
    }
}

extern "C" void kernel_launch(void* const* d_in, const int* in_sizes, int n_in,
                              void* d_out, int out_size, void* d_ws, size_t ws_size,
                              hipStream_t stream) {
    const float* x     = (const float*)d_in[0];  // [B, 256]
    const float* W_lin = (const float*)d_in[1];  // [256, 256]
    const float* b_lin = (const float*)d_in[2];  // [256]
    // d_in[3] (W_sel) and d_in[4] (b_sel) are algebraically dead: mixed == 0.
    float* out = (float*)d_out;

    const int B = in_sizes[0] / N_IN;            // 8192

    // Kernel 1: live GEMM + threshold into cols [0,256).
    dim3 gemm_grid(2, B / 16);
    snn_gemm_heaviside<<<gemm_grid, 256, 0, stream>>>(x, W_lin, b_lin, out);

    // Kernel 2: exact zeros into cols [256, 8448).  ~277 MB, HBM-bound.
    const long long total4 = (long long)B * 2048;
    snn_zero_tail<<<4096, 256, 0, stream>>>((float4*)out, total4);
}